// GraphSAGE_JStar_Predictor_82609400971334
// MI455X (gfx1250) — compile-verified
//
#include <hip/hip_runtime.h>
#include <math.h>

typedef __attribute__((ext_vector_type(2))) float v2f;
typedef __attribute__((ext_vector_type(8))) float v8f;

#define N_NODES  100000
#define N_EDGES  1600000
#define N_GRAPHS 1000
#define F_IN     9
#define HID      64

// ---------------- utility ----------------
__global__ void zero_kernel(float* __restrict__ p, long long n) {
  long long i = (long long)blockIdx.x * blockDim.x + threadIdx.x;
  if (i < n) p[i] = 0.0f;
}

__global__ void degree_kernel(const int* __restrict__ dst, float* __restrict__ cnt, int E) {
  int e = blockIdx.x * blockDim.x + threadIdx.x;
  if (e < E) atomicAdd(&cnt[dst[e]], 1.0f);
}

__global__ void gcount_kernel(const int* __restrict__ batch, float* __restrict__ gcnt, int n) {
  int i = blockIdx.x * blockDim.x + threadIdx.x;
  if (i < n) atomicAdd(&gcnt[batch[i]], 1.0f);
}

// ---------------- layer 1: scatter (E x 9) ----------------
__global__ void scatter_f9(const float* __restrict__ x, const int* __restrict__ src,
                           const int* __restrict__ dst, float* __restrict__ s1, int E) {
  int e = blockIdx.x * blockDim.x + threadIdx.x;
  int f = blockIdx.y;            // 0..8
  if (e < E) {
    int s = src[e], d = dst[e];
    atomicAdd(&s1[d * F_IN + f], x[s * F_IN + f]);
  }
}

// ---------------- layer 1: node update (K=9, VALU) ----------------
__global__ void layer1_kernel(const float* __restrict__ s1, const float* __restrict__ cnt,
                              const float* __restrict__ x, const float* __restrict__ Wl,
                              const float* __restrict__ Wr, const float* __restrict__ b,
                              float* __restrict__ h1) {
  __shared__ float sa[F_IN], sx[F_IN];
  int n = blockIdx.x;
  int j = threadIdx.x;           // 0..63 output feature
  if (j < F_IN) {
    float inv = 1.0f / fmaxf(cnt[n], 1.0f);
    sa[j] = s1[n * F_IN + j] * inv;
    sx[j] = x[n * F_IN + j];
  }
  __syncthreads();
  float acc = b[j];
  #pragma unroll
  for (int k = 0; k < F_IN; ++k)
    acc = fmaf(sa[k], Wl[k * HID + j], fmaf(sx[k], Wr[k * HID + j], acc));
  h1[(long long)n * HID + j] = fmaxf(acc, 0.0f);   // relu
}

// ---------------- layer 2: scatter (E x 64) ----------------
__global__ void scatter_f64(const float* __restrict__ h1, const int* __restrict__ src,
                            const int* __restrict__ dst, float* __restrict__ s2, long long total) {
  long long i = (long long)blockIdx.x * blockDim.x + threadIdx.x;
  if (i < total) {
    int e = (int)(i >> 6);
    int f = (int)(i & 63);
    int s = src[e], d = dst[e];
    atomicAdd(&s2[(long long)d * HID + f], h1[(long long)s * HID + f]);
  }
}

// ---------------- layer 2: WMMA node update fused with graph mean-pool ----------------
// D = (s2/cnt) @ Wl2 + h1 @ Wr2 + b2 ; then atomicAdd into gsum[batch[node]]
// 4 waves per block; wave w owns output columns [16w, 16w+16). fp32 WMMA 16x16x4.
__global__ void layer2_wmma_kernel(const float* __restrict__ s2, const float* __restrict__ cnt,
                                   const float* __restrict__ h1, const float* __restrict__ Wl,
                                   const float* __restrict__ Wr, const float* __restrict__ b,
                                   const int* __restrict__ batch, float* __restrict__ gsum,
                                   int nNodes) {
  __shared__ float lds_a[16][68];   // mean-aggregated neighbor feats (padded: no bank conflict)
  __shared__ float lds_h[16][68];   // root feats
  __shared__ float inv_cnt[16];

  int tid  = threadIdx.x;
  int row0 = blockIdx.x * 16;

  if (tid < 16) {
    int node = row0 + tid;
    inv_cnt[tid] = (node < nNodes) ? (1.0f / fmaxf(cnt[node], 1.0f)) : 0.0f;
  }
  __syncthreads();

  // stage 16x64 tiles: 256 float4 chunks, 128 threads x 2
  #pragma unroll
  for (int v = tid; v < 256; v += 128) {
    int row = v >> 4;              // 16 float4 per row
    int c4  = v & 15;
    int node = row0 + row;
    float4 a, h;
    if (node < nNodes) {
      const float4* s2p = (const float4*)(s2 + (long long)node * HID);
      const float4* h1p = (const float4*)(h1 + (long long)node * HID);
      a = s2p[c4]; h = h1p[c4];
      float inv = inv_cnt[row];
      a.x *= inv; a.y *= inv; a.z *= inv; a.w *= inv;
    } else {
      a = make_float4(0.f, 0.f, 0.f, 0.f); h = a;
    }
    int col = c4 * 4;
    lds_a[row][col] = a.x; lds_a[row][col+1] = a.y; lds_a[row][col+2] = a.z; lds_a[row][col+3] = a.w;
    lds_h[row][col] = h.x; lds_h[row][col+1] = h.y; lds_h[row][col+2] = h.z; lds_h[row][col+3] = h.w;
  }
  __syncthreads();

  int lane = tid & 31;
  int wv   = tid >> 5;             // 0..3 -> output column slice
  int half = (lane >> 4) & 1;      // lanes 16-31 hold K+2,K+3 / M+8
  int lo   = lane & 15;
  int colb = wv * 16;

  v8f c = {};
  #pragma unroll
  for (int k = 0; k < HID; k += 4) {
    int ka = k + 2 * half;
    v2f aagg, ah, bl, br;
    aagg[0] = lds_a[lo][ka];  aagg[1] = lds_a[lo][ka + 1];
    ah[0]   = lds_h[lo][ka];  ah[1]   = lds_h[lo][ka + 1];
    bl[0] = Wl[ka * HID + colb + lo];  bl[1] = Wl[(ka + 1) * HID + colb + lo];
    br[0] = Wr[ka * HID + colb + lo];  br[1] = Wr[(ka + 1) * HID + colb + lo];
    c = __builtin_amdgcn_wmma_f32_16x16x4_f32(false, aagg, false, bl, (short)0, c, false, false);
    c = __builtin_amdgcn_wmma_f32_16x16x4_f32(false, ah,   false, br, (short)0, c, false, false);
  }

  // epilogue: bias + fused global-mean-pool accumulation (no relu after last conv)
  int col   = colb + lo;
  float bia = b[col];
  #pragma unroll
  for (int v = 0; v < 8; ++v) {
    int m = v + 8 * half;          // C/D layout: M = v + 8*(lane>=16), N = lo
    int node = row0 + m;
    if (node < nNodes) {
      float val = c[v] + bia;
      atomicAdd(&gsum[(long long)batch[node] * HID + col], val);
    }
  }
}

// ---------------- readout MLP: 64 -> 32 relu -> 1 sigmoid ----------------
__global__ void readout_kernel(const float* __restrict__ gsum, const float* __restrict__ gcnt,
                               const float* __restrict__ Wm1, const float* __restrict__ bm1,
                               const float* __restrict__ Wm2, const float* __restrict__ bm2,
                               float* __restrict__ out, int G) {
  int g = blockIdx.x * blockDim.x + threadIdx.x;
  if (g >= G) return;
  float inv = 1.0f / fmaxf(gcnt[g], 1.0f);
  float xg[HID];
  #pragma unroll
  for (int f = 0; f < HID; ++f) xg[f] = gsum[(long long)g * HID + f] * inv;
  float z = bm2[0];
  #pragma unroll 4
  for (int j = 0; j < 32; ++j) {
    float a = bm1[j];
    #pragma unroll
    for (int f = 0; f < HID; ++f) a = fmaf(xg[f], Wm1[f * 32 + j], a);
    a = fmaxf(a, 0.0f);
    z = fmaf(a, Wm2[j], z);
  }
  out[g] = 1.0f / (1.0f + expf(-z));
}

// ---------------- launch ----------------
extern "C" void kernel_launch(void* const* d_in, const int* in_sizes, int n_in,
                              void* d_out, int out_size, void* d_ws, size_t ws_size,
                              hipStream_t stream) {
  const float* x    = (const float*)d_in[0];
  const int*   ei   = (const int*)  d_in[1];   // [2, E]
  const int*   batch= (const int*)  d_in[3];
  const float* Wl1  = (const float*)d_in[4];
  const float* Wr1  = (const float*)d_in[5];
  const float* b1   = (const float*)d_in[6];
  const float* Wl2  = (const float*)d_in[7];
  const float* Wr2  = (const float*)d_in[8];
  const float* b2   = (const float*)d_in[9];
  const float* Wm1  = (const float*)d_in[10];
  const float* bm1  = (const float*)d_in[11];
  const float* Wm2  = (const float*)d_in[12];
  const float* bm2  = (const float*)d_in[13];
  const int* src  = ei;
  const int* dstE = ei + N_EDGES;

  // workspace carve (floats): cnt[N] | s1[9N] | s2[64N] | gsum[64G] | gcnt[G] | h1[64N]
  float* ws   = (float*)d_ws;
  float* cnt  = ws;
  float* s1   = cnt  + N_NODES;
  float* s2   = s1   + 9LL  * N_NODES;
  float* gsum = s2   + 64LL * N_NODES;
  float* gcnt = gsum + 64LL * N_GRAPHS;
  float* h1   = gcnt + N_GRAPHS;

  long long zero_n = 74LL * N_NODES + 65LL * N_GRAPHS;   // everything except h1
  zero_kernel<<<(int)((zero_n + 1023) / 1024), 1024, 0, stream>>>(ws, zero_n);

  degree_kernel<<<(N_EDGES + 255) / 256, 256, 0, stream>>>(dstE, cnt, N_EDGES);
  gcount_kernel<<<(N_NODES + 255) / 256, 256, 0, stream>>>(batch, gcnt, N_NODES);

  scatter_f9<<<dim3((N_EDGES + 255) / 256, F_IN), 256, 0, stream>>>(x, src, dstE, s1, N_EDGES);
  layer1_kernel<<<N_NODES, HID, 0, stream>>>(s1, cnt, x, Wl1, Wr1, b1, h1);

  long long tot2 = (long long)N_EDGES * HID;
  scatter_f64<<<(int)((tot2 + 255) / 256), 256, 0, stream>>>(h1, src, dstE, s2, tot2);

  layer2_wmma_kernel<<<(N_NODES + 15) / 16, 128, 0, stream>>>(s2, cnt, h1, Wl2, Wr2, b2,
                                                              batch, gsum, N_NODES);

  readout_kernel<<<(N_GRAPHS + 127) / 128, 128, 0, stream>>>(gsum, gcnt, Wm1, bm1, Wm2, bm2,
                                                             (float*)d_out, N_GRAPHS);
}